// cosine_regularizer_33956011442258
// MI455X (gfx1250) — compile-verified
//
#include <hip/hip_runtime.h>

typedef __attribute__((ext_vector_type(2))) float v2f;
typedef __attribute__((ext_vector_type(8))) float v8f;

#define N_ROWS 16384
#define D_COLS 256
#define TILE_ROWS 32
#define TILES_PER_BLOCK 2
#define NBLOCKS (N_ROWS / (TILE_ROWS * TILES_PER_BLOCK))   // 256
#define LDS_STRIDE 272   // 256 + 16 floats pad: rows 2 apart land in disjoint bank sets

// Kernel 1: per-block partial column sums of the row-normalized matrix,
// accumulated on the matrix pipe via V_WMMA_F32_16X16X4_F32 with A = ones.
__global__ __launch_bounds__(256) void cosreg_colsum_kernel(
    const float* __restrict__ points, float* __restrict__ partial) {
  __shared__ float tile[TILE_ROWS * LDS_STRIDE];
  __shared__ float inv_norm[TILE_ROWS];

  const int tid  = threadIdx.x;
  const int wave = tid >> 5;
  const int lane = tid & 31;
  const int colN = lane & 15;   // N within the 16-wide column slice
  const int half = lane >> 4;   // 0 -> K rows {0,1}, 1 -> K rows {2,3}

  const int c0a = wave * 16;          // slice A: cols   0..127 over 8 waves
  const int c0b = wave * 16 + 128;    // slice B: cols 128..255 over 8 waves

  v8f accA = {};
  v8f accB = {};
  const v2f ones = {1.0f, 1.0f};

  for (int t = 0; t < TILES_PER_BLOCK; ++t) {
    const int row0 = (blockIdx.x * TILES_PER_BLOCK + t) * TILE_ROWS;

    // ---- coalesced float4 load of 32x256 tile into padded LDS ----
    {
      const float4* __restrict__ src =
          (const float4*)(points + (size_t)row0 * D_COLS);
      const int r  = tid >> 6;   // 0..3
      const int c4 = tid & 63;   // 0..63 (float4 column)
#pragma unroll
      for (int i = 0; i < TILE_ROWS / 4; ++i) {
        const int rr = i * 4 + r;
        float4 v = src[(size_t)rr * (D_COLS / 4) + c4];
        float* dst = &tile[rr * LDS_STRIDE + c4 * 4];
        dst[0] = v.x; dst[1] = v.y; dst[2] = v.z; dst[3] = v.w;
      }
    }
    __syncthreads();

    // ---- per-row 1/||row||: wave w reduces rows w*4 .. w*4+3 ----
#pragma unroll
    for (int rr = 0; rr < TILE_ROWS / 8; ++rr) {
      const int r = wave * (TILE_ROWS / 8) + rr;
      const float* __restrict__ rowp = &tile[r * LDS_STRIDE];
      float s = 0.0f;
#pragma unroll
      for (int j = 0; j < D_COLS / 32; ++j) {
        float x = rowp[lane + 32 * j];
        s += x * x;
      }
#pragma unroll
      for (int off = 16; off >= 1; off >>= 1) s += __shfl_xor(s, off, 32);
      if (lane == 0) inv_norm[r] = 1.0f / sqrtf(s);
    }
    __syncthreads();

    // ---- WMMA column-sum accumulation: D = ones(16x4) x B(4x16) + C ----
#pragma unroll
    for (int k0 = 0; k0 < TILE_ROWS; k0 += 4) {
      const int r0 = k0 + half * 2;
      const float in0 = inv_norm[r0];
      const float in1 = inv_norm[r0 + 1];
      v2f bA, bB;
      bA[0] = tile[r0 * LDS_STRIDE + c0a + colN] * in0;
      bA[1] = tile[(r0 + 1) * LDS_STRIDE + c0a + colN] * in1;
      bB[0] = tile[r0 * LDS_STRIDE + c0b + colN] * in0;
      bB[1] = tile[(r0 + 1) * LDS_STRIDE + c0b + colN] * in1;
      accA = __builtin_amdgcn_wmma_f32_16x16x4_f32(
          false, ones, false, bA, (short)0, accA, false, false);
      accB = __builtin_amdgcn_wmma_f32_16x16x4_f32(
          false, ones, false, bB, (short)0, accB, false, false);
    }
    __syncthreads();  // done reading tile before next iteration overwrites it
  }

  // D-matrix VGPR0, lanes 0..15 = (M=0, N=lane): one copy of the column sums.
  if (lane < 16) {
    float* __restrict__ outp = partial + (size_t)blockIdx.x * D_COLS;
    outp[c0a + lane] = accA[0];
    outp[c0b + lane] = accB[0];
  }
}

// Kernel 2: s[c] = sum_b partial[b][c] (fixed order -> deterministic),
// then (||s||^2 - N) / N^2.
__global__ __launch_bounds__(256) void cosreg_finish_kernel(
    const float* __restrict__ partial, float* __restrict__ out) {
  __shared__ float red[256];
  const int t = threadIdx.x;
  float s = 0.0f;
  for (int b = 0; b < NBLOCKS; ++b) s += partial[(size_t)b * D_COLS + t];
  red[t] = s * s;
  __syncthreads();
#pragma unroll
  for (int off = 128; off >= 1; off >>= 1) {
    if (t < off) red[t] += red[t + off];
    __syncthreads();
  }
  if (t == 0) {
    const float n = (float)N_ROWS;
    out[0] = (red[0] - n) / (n * n);
  }
}

extern "C" void kernel_launch(void* const* d_in, const int* in_sizes, int n_in,
                              void* d_out, int out_size, void* d_ws, size_t ws_size,
                              hipStream_t stream) {
  (void)in_sizes; (void)n_in; (void)out_size; (void)ws_size;
  const float* points = (const float*)d_in[0];
  float* out = (float*)d_out;
  float* partial = (float*)d_ws;  // NBLOCKS * D_COLS floats = 256 KB

  cosreg_colsum_kernel<<<NBLOCKS, 256, 0, stream>>>(points, partial);
  cosreg_finish_kernel<<<1, 256, 0, stream>>>(partial, out);
}